// Encoder_33105607917952
// MI455X (gfx1250) — compile-verified
//
#include <hip/hip_runtime.h>
#include <hip/hip_bf16.h>

#define N_NODES   100000
#define N_EDGES   3200000
#define IN_FEAT   256
#define OUT_FEAT  128

typedef __attribute__((ext_vector_type(2))) float v2f;
typedef __attribute__((ext_vector_type(8))) float v8f;

// ---------------------------------------------------------------------------
// Kernel 1: zero-init the SpMM accumulator region (x) of d_out.
// ---------------------------------------------------------------------------
__global__ void zero_f4_kernel(float4* __restrict__ p, int n4) {
    int i = blockIdx.x * blockDim.x + threadIdx.x;
    if (i < n4) {
        p[i] = make_float4(0.f, 0.f, 0.f, 0.f);
    }
}

// ---------------------------------------------------------------------------
// Kernel 2: feat_embedding = feat @ W  using V_WMMA_F32_16X16X4_F32.
// One wave (32 lanes) computes one 16x16 output tile. 8 waves / block.
// W (256x128 f32 = 128 KB) is staged once per block into LDS (CDNA5 WGP has
// 320 KB LDS), so B-operand reads are ds_loads shared by all 8 waves.
//
// WMMA f32 16x16x4 VGPR layouts (ISA 7.12.2):
//   A (16x4):  lanes 0-15: M=lane, v0=K0,v1=K1 ; lanes 16-31: M=lane-16, v0=K2,v1=K3
//   B (4x16):  lanes 0-15: N=lane, v0=K0,v1=K1 ; lanes 16-31: N=lane-16, v0=K2,v1=K3
//   C/D:       VGPR r: lanes 0-15 -> M=r, N=lane ; lanes 16-31 -> M=r+8, N=lane-16
// ---------------------------------------------------------------------------
__launch_bounds__(256)
__global__ void gemm_wmma_f32_kernel(const float* __restrict__ feat,
                                     const float* __restrict__ W,
                                     float* __restrict__ femb) {
    __shared__ float sW[IN_FEAT * OUT_FEAT];   // 128 KB LDS

    // Cooperative load of W into LDS (float4 vectorized).
    {
        const float4* Wv  = (const float4*)W;
        float4*       sWv = (float4*)sW;
        const int n4 = IN_FEAT * OUT_FEAT / 4;       // 8192
        for (int i = threadIdx.x; i < n4; i += blockDim.x) {
            sWv[i] = Wv[i];
        }
    }
    __syncthreads();

    const int lane   = threadIdx.x & 31;
    const int waveId = threadIdx.x >> 5;            // 0..7
    const int tile   = blockIdx.x * 8 + waveId;     // 50000 tiles total
    const int tileM  = tile >> 3;                   // 0..6249
    const int tileN  = tile & 7;                    // 0..7

    const int row0 = tileM * 16;
    const int n0   = tileN * 16;
    if (row0 >= N_NODES) return;

    const int m    = lane & 15;
    const int half = lane >> 4;                     // 0: K0/K1, 1: K2/K3
    const int nn   = lane & 15;

    const float* arow = feat + (size_t)(row0 + m) * IN_FEAT;

    v8f c = {};
    #pragma unroll 4
    for (int k = 0; k < IN_FEAT; k += 4) {
        // A operand: two consecutive K values for this lane (8B load).
        float2 a2 = *(const float2*)(arow + k + 2 * half);
        v2f a; a[0] = a2.x; a[1] = a2.y;
        // B operand from LDS: rows k+2*half, k+2*half+1, column n0+nn.
        v2f b;
        b[0] = sW[(k + 2 * half    ) * OUT_FEAT + n0 + nn];
        b[1] = sW[(k + 2 * half + 1) * OUT_FEAT + n0 + nn];
        // (neg_a, A, neg_b, B, c_mod, C, reuse_a, reuse_b)
        c = __builtin_amdgcn_wmma_f32_16x16x4_f32(
                false, a, false, b, (short)0, c, false, false);
    }

    // Store 16x16 tile: element c[r] -> row (row0 + r + 8*half), col (n0 + nn).
    float* orow = femb + (size_t)(row0 + half * 8) * OUT_FEAT + n0 + nn;
    #pragma unroll
    for (int r = 0; r < 8; ++r) {
        orow[(size_t)r * OUT_FEAT] = c[r];
    }
}

// ---------------------------------------------------------------------------
// Kernel 3: SpMM with sorted rows.
// Each wave owns EDGES_PER_WAVE contiguous edges; each lane owns 4 of the 128
// features (float4). Register-accumulate while row id is constant (rows are
// sorted), flush with global_atomic_add_f32 only at segment boundaries.
// Edge metadata is loaded 32-wide per lane and broadcast via wave32 shuffles.
// femb (51 MB) is L2-resident on MI455X (192 MB L2), so gathers hit L2.
// ---------------------------------------------------------------------------
#define EDGES_PER_WAVE 256

__device__ __forceinline__ void flush_row(float* __restrict__ x, int rowi,
                                          int lane, const float4& acc) {
    float* p = x + (size_t)rowi * OUT_FEAT + lane * 4;
    atomicAdd(p + 0, acc.x);
    atomicAdd(p + 1, acc.y);
    atomicAdd(p + 2, acc.z);
    atomicAdd(p + 3, acc.w);
}

__launch_bounds__(256)
__global__ void spmm_kernel(const int*   __restrict__ erow,
                            const int*   __restrict__ ecol,
                            const float* __restrict__ eval,
                            const float* __restrict__ femb,
                            float*       __restrict__ x) {
    const int lane = threadIdx.x & 31;
    const int wid  = (blockIdx.x * blockDim.x + threadIdx.x) >> 5;

    int e0 = wid * EDGES_PER_WAVE;
    if (e0 >= N_EDGES) return;
    int e1 = e0 + EDGES_PER_WAVE;
    if (e1 > N_EDGES) e1 = N_EDGES;

    const float4* fembv = (const float4*)femb;   // 32 float4 per embedding row

    float4 acc = make_float4(0.f, 0.f, 0.f, 0.f);
    int cur = -1;                                // no open segment yet

    for (int eb = e0; eb < e1; eb += 32) {
        const int e = eb + lane;
        const bool ok = (e < e1);
        int   r_l = ok ? erow[e] : -1;
        int   c_l = ok ? ecol[e] : 0;
        float v_l = ok ? eval[e] : 0.f;

        #pragma unroll 8
        for (int j = 0; j < 32; ++j) {
            const int r = __shfl(r_l, j, 32);
            if (r < 0) break;                    // past end; uniform across wave
            const int   col = __shfl(c_l, j, 32);
            const float v   = __shfl(v_l, j, 32);

            if (r != cur) {
                if (cur >= 0) flush_row(x, cur, lane, acc);
                acc = make_float4(0.f, 0.f, 0.f, 0.f);
                cur = r;
            }
            const float4 f = fembv[(size_t)col * (OUT_FEAT / 4) + lane];
            acc.x = fmaf(v, f.x, acc.x);
            acc.y = fmaf(v, f.y, acc.y);
            acc.z = fmaf(v, f.z, acc.z);
            acc.w = fmaf(v, f.w, acc.w);
        }
    }
    if (cur >= 0) flush_row(x, cur, lane, acc);
}

// ---------------------------------------------------------------------------
// Launch: zero(x) -> GEMM -> SpMM, all on `stream` (stream order gives the
// GEMM->SpMM dependency).
// ---------------------------------------------------------------------------
extern "C" void kernel_launch(void* const* d_in, const int* in_sizes, int n_in,
                              void* d_out, int out_size, void* d_ws, size_t ws_size,
                              hipStream_t stream) {
    const float* feat  = (const float*)d_in[0];
    const int*   erow  = (const int*)  d_in[1];
    const int*   ecol  = (const int*)  d_in[2];
    const float* eval  = (const float*)d_in[3];
    const float* W     = (const float*)d_in[4];

    float* femb = (float*)d_out;                                  // [N, 128]
    float* x    = (float*)d_out + (size_t)N_NODES * OUT_FEAT;     // [N, 128]

    // 1) zero the accumulator region.
    {
        const int n4 = N_NODES * OUT_FEAT / 4;                    // 3.2M float4
        const int blocks = (n4 + 255) / 256;
        zero_f4_kernel<<<blocks, 256, 0, stream>>>((float4*)x, n4);
    }

    // 2) dense GEMM via WMMA: 6250 blocks x 8 waves = 50000 16x16 tiles.
    {
        const int tilesM = N_NODES / 16;                          // 6250
        gemm_wmma_f32_kernel<<<tilesM, 256, 0, stream>>>(feat, W, femb);
    }

    // 3) SpMM: 3.2M edges / (256 edges/wave * 8 waves/block) = 1563 blocks.
    {
        const int waves  = (N_EDGES + EDGES_PER_WAVE - 1) / EDGES_PER_WAVE;
        const int blocks = (waves + 7) / 8;
        spmm_kernel<<<blocks, 256, 0, stream>>>(erow, ecol, eval, femb, x);
    }
}